// Net_for_classification3_61357902791131
// MI455X (gfx1250) — compile-verified
//
#include <hip/hip_runtime.h>
#include <hip/hip_bf16.h>

typedef __attribute__((ext_vector_type(2))) float v2f;
typedef __attribute__((ext_vector_type(8))) float v8f;

#define FD 64          // feature dim
#define NCLS 10
#define NGRAPH 256

// ---------------------------------------------------------------------------
// H[N x 64] = X[N x 64] @ W[64 x 64]   via v_wmma_f32_16x16x4_f32
// grid.x = N/16 row tiles, 128 threads = 4 waves, wave w owns N-cols [16w,16w+16)
// ---------------------------------------------------------------------------
__global__ __launch_bounds__(128) void gcn_gemm_wmma(
    const float* __restrict__ X, const float* __restrict__ W,
    float* __restrict__ H, int nrows) {
  const int lane = threadIdx.x & 31;
  const int wave = threadIdx.x >> 5;
  const int m0 = blockIdx.x * 16;
  if (m0 >= nrows) return;
  const int n0 = wave * 16;
  const int half = lane >> 4;     // 0 = lanes 0-15, 1 = lanes 16-31
  const int lm = lane & 15;
  const int kb = half * 2;        // K sub-offset per ISA A/B layout

  const float* xr = X + (size_t)(m0 + lm) * FD + kb;      // A: row m0+lm
  const float* wp = W + (size_t)kb * FD + (n0 + lm);      // B: col n0+lm

  v8f acc = {};
#pragma unroll
  for (int kk = 0; kk < FD; kk += 4) {
    v2f a, b;
    a.x = xr[kk];
    a.y = xr[kk + 1];
    b.x = wp[kk * FD];
    b.y = wp[kk * FD + FD];
    acc = __builtin_amdgcn_wmma_f32_16x16x4_f32(
        /*neg_a=*/false, a, /*neg_b=*/false, b,
        /*c_mod=*/(short)0, acc, /*reuse_a=*/false, /*reuse_b=*/false);
  }
#pragma unroll
  for (int r = 0; r < 8; ++r) {
    int row = m0 + r + half * 8;          // D layout: lanes0-15 rows 0-7, lanes16-31 rows 8-15
    H[(size_t)row * FD + n0 + lm] = acc[r];
  }
}

// ---------------------------------------------------------------------------
// helpers
// ---------------------------------------------------------------------------
__global__ void k_set1(float* __restrict__ p, int n) {
  int i = blockIdx.x * blockDim.x + threadIdx.x;
  if (i < n) p[i] = 1.0f;
}

__global__ void k_zero(float* __restrict__ p, int n) {
  int i = blockIdx.x * blockDim.x + threadIdx.x;
  if (i < n) p[i] = 0.0f;
}

__global__ void k_degree(const int* __restrict__ dst, float* __restrict__ deg, int E) {
  int e = blockIdx.x * blockDim.x + threadIdx.x;
  if (e < E) atomicAdd(&deg[dst[e]], 1.0f);
}

__global__ void k_rsqrt(float* __restrict__ deg, int n) {
  int i = blockIdx.x * blockDim.x + threadIdx.x;
  if (i < n) deg[i] = rsqrtf(deg[i]);     // deg >= 1 (self loop) -> safe
}

// O[i][f] = H[i][f]*dinv[i]^2 + b[f]   (self-loop term + bias, also inits O)
__global__ void k_selfloop_bias(const float* __restrict__ H,
                                const float* __restrict__ dinv,
                                const float* __restrict__ b,
                                float* __restrict__ O, int nnodes) {
  int idx = blockIdx.x * blockDim.x + threadIdx.x;
  if (idx >= nnodes * FD) return;
  int i = idx >> 6;
  int f = idx & 63;
  float di = dinv[i];
  O[idx] = H[idx] * di * di + b[f];
}

// edge scatter: O[dst] += H[src] * dinv[src]*dinv[dst]; 64 lanes per edge (coalesced)
__global__ __launch_bounds__(256) void k_edges(
    const int* __restrict__ src, const int* __restrict__ dst,
    const float* __restrict__ dinv, const float* __restrict__ H,
    float* __restrict__ O, int E) {
  long long t = (long long)blockIdx.x * 256 + threadIdx.x;
  int e = (int)(t >> 6);
  int f = (int)(t & 63);
  if (e >= E) return;
  int s = src[e];
  int d = dst[e];
  float norm = dinv[s] * dinv[d];
  atomicAdd(&O[(size_t)d * FD + f], H[(size_t)s * FD + f] * norm);
}

__global__ void k_relu(float* __restrict__ O, int n) {
  int i = blockIdx.x * blockDim.x + threadIdx.x;
  if (i < n) O[i] = fmaxf(O[i], 0.0f);
}

// mean-pool accumulation
__global__ void k_pool(const float* __restrict__ O, const int* __restrict__ batch,
                       float* __restrict__ sums, float* __restrict__ cnt, int nnodes) {
  int idx = blockIdx.x * blockDim.x + threadIdx.x;
  if (idx >= nnodes * FD) return;
  int i = idx >> 6;
  int f = idx & 63;
  int g = batch[i];
  atomicAdd(&sums[(size_t)g * FD + f], O[idx]);
  if (f == 0) atomicAdd(&cnt[g], 1.0f);
}

// out[g][c] = (sums[g]/max(cnt,1)) . Wfc[:,c] + bfc[c]
__global__ void k_fc(const float* __restrict__ sums, const float* __restrict__ cnt,
                     const float* __restrict__ Wfc, const float* __restrict__ bfc,
                     float* __restrict__ out) {
  int idx = blockIdx.x * blockDim.x + threadIdx.x;
  if (idx >= NGRAPH * NCLS) return;
  int g = idx / NCLS;
  int c = idx % NCLS;
  float inv = 1.0f / fmaxf(cnt[g], 1.0f);
  float acc = bfc[c];
  const float* row = sums + (size_t)g * FD;
#pragma unroll
  for (int k = 0; k < FD; ++k) acc += row[k] * inv * Wfc[k * NCLS + c];
  out[idx] = acc;
}

// ---------------------------------------------------------------------------
static inline size_t align256(size_t x) { return (x + 255) & ~(size_t)255; }

extern "C" void kernel_launch(void* const* d_in, const int* in_sizes, int n_in,
                              void* d_out, int out_size, void* d_ws, size_t ws_size,
                              hipStream_t stream) {
  const float* x   = (const float*)d_in[0];
  const int*  ei   = (const int*)d_in[1];
  const int*  batch= (const int*)d_in[2];
  const float* W1  = (const float*)d_in[3];
  const float* b1  = (const float*)d_in[4];
  const float* W2  = (const float*)d_in[5];
  const float* b2  = (const float*)d_in[6];
  const float* W3  = (const float*)d_in[7];
  const float* b3  = (const float*)d_in[8];
  const float* Wfc = (const float*)d_in[9];
  const float* bfc = (const float*)d_in[10];
  float* out = (float*)d_out;

  const int N = in_sizes[0] / FD;       // 50000
  const int E = in_sizes[1] / 2;        // 800000
  const int* src = ei;
  const int* dst = ei + E;

  // workspace carve-out
  char* ws = (char*)d_ws;
  size_t off = 0;
  float* dinv = (float*)(ws + off); off += align256((size_t)N * sizeof(float));
  float* H    = (float*)(ws + off); off += align256((size_t)N * FD * sizeof(float));
  float* O    = (float*)(ws + off); off += align256((size_t)N * FD * sizeof(float));
  float* sums = (float*)(ws + off); off += align256((size_t)NGRAPH * FD * sizeof(float));
  float* cnt  = (float*)(ws + off); off += align256((size_t)NGRAPH * sizeof(float));
  (void)ws_size;

  const int nf  = N * FD;
  const int gN   = (N + 255) / 256;
  const int gNF  = (nf + 255) / 256;
  const int gE   = (E + 255) / 256;
  const int gE64 = (int)(((long long)E * FD + 255) / 256);
  const int gGemm = N / 16;             // N divisible by 16

  // degrees -> dinv (shared by all layers)
  k_set1<<<gN, 256, 0, stream>>>(dinv, N);
  k_degree<<<gE, 256, 0, stream>>>(dst, dinv, E);
  k_rsqrt<<<gN, 256, 0, stream>>>(dinv, N);

  // layer 1
  gcn_gemm_wmma<<<gGemm, 128, 0, stream>>>(x, W1, H, N);
  k_selfloop_bias<<<gNF, 256, 0, stream>>>(H, dinv, b1, O, N);
  k_edges<<<gE64, 256, 0, stream>>>(src, dst, dinv, H, O, E);
  k_relu<<<gNF, 256, 0, stream>>>(O, nf);

  // layer 2
  gcn_gemm_wmma<<<gGemm, 128, 0, stream>>>(O, W2, H, N);
  k_selfloop_bias<<<gNF, 256, 0, stream>>>(H, dinv, b2, O, N);
  k_edges<<<gE64, 256, 0, stream>>>(src, dst, dinv, H, O, E);
  k_relu<<<gNF, 256, 0, stream>>>(O, nf);

  // layer 3 (no relu)
  gcn_gemm_wmma<<<gGemm, 128, 0, stream>>>(O, W3, H, N);
  k_selfloop_bias<<<gNF, 256, 0, stream>>>(H, dinv, b3, O, N);
  k_edges<<<gE64, 256, 0, stream>>>(src, dst, dinv, H, O, E);

  // mean pool + fc
  k_zero<<<(NGRAPH * FD + 255) / 256, 256, 0, stream>>>(sums, NGRAPH * FD);
  k_zero<<<1, 256, 0, stream>>>(cnt, NGRAPH);
  k_pool<<<gNF, 256, 0, stream>>>(O, batch, sums, cnt, N);
  k_fc<<<(NGRAPH * NCLS + 255) / 256, 256, 0, stream>>>(sums, cnt, Wfc, bfc, out);
}